// S4Layer_34660386079078
// MI455X (gfx1250) — compile-verified
//
#include <hip/hip_runtime.h>
#include <hip/hip_bf16.h>

// ---------------- problem constants (from reference setup_inputs) ----------
#define BATCH 8
#define SEQ   4096
#define DM    1024
#define DS    16
#define CH_L  16                 // chunk length (== WMMA tile M/N/K-group)
#define NCH   (SEQ / CH_L)       // 256 chunks per (b,d)
#define NCOL  (BATCH * NCH)      // 2048 matmul columns per channel d
#define MATS_PER_D 784           // K(256) + Bm(256) + Cm(256) + lamL(16)

typedef float v2f __attribute__((ext_vector_type(2)));
typedef float v8f __attribute__((ext_vector_type(8)));

// LDS padding strides (bank-conflict free on 64 banks)
#define SDX 276   // per-d stride for 16x16 tiles with inner row stride 17
#define SDM 288   // per-d stride for 16x(16->18 padded) matrices

// ---------------------------------------------------------------------------
// Phase 0: per-channel chunk matrices.
//   lam_n = 1 - exp(A_log[d,n]);  k[tau] = sum_n C_n B_n lam_n^tau
//   K[t][s]  = k[t-s] (t>=s) + D on diagonal
//   Bm[n][s] = B_n lam_n^(15-s)
//   Cm[t][n] = C_n lam_n^(t+1)
//   lamL[n]  = lam_n^16
// ---------------------------------------------------------------------------
__global__ void s4_prep(const float* __restrict__ A_log, const float* __restrict__ B,
                        const float* __restrict__ C, const float* __restrict__ D,
                        float* __restrict__ mats) {
  int d = blockIdx.x;
  int t = threadIdx.x;                      // 0..31 (one wave)
  __shared__ float Bv[DS], Cv[DS], CB[DS];
  __shared__ float pw[DS][CH_L + 1];
  __shared__ float kv[CH_L];
  if (t < DS) {
    float l = 1.0f - __expf(A_log[d * DS + t]);
    Bv[t] = B[d * DS + t];
    Cv[t] = C[d * DS + t];
    CB[t] = Bv[t] * Cv[t];
    float p = 1.0f;
    for (int j = 0; j <= CH_L; ++j) { pw[t][j] = p; p *= l; }
  }
  __syncthreads();
  if (t < CH_L) {
    float s = 0.0f;
    for (int n = 0; n < DS; ++n) s += CB[n] * pw[n][t];
    kv[t] = s;
  }
  __syncthreads();
  float* K    = mats + (size_t)d * MATS_PER_D;
  float* Bm   = K + 256;
  float* Cm   = K + 512;
  float* lamL = K + 768;
  float Dd = D[d];
  for (int i = t; i < 256; i += 32) {
    int r = i >> 4, c = i & 15;
    float kval = (r >= c) ? kv[r - c] : 0.0f;
    if (r == c) kval += Dd;
    K[i]  = kval;
    Bm[i] = Bv[r] * pw[r][15 - c];
    Cm[i] = Cv[c] * pw[c][r + 1];
  }
  if (t < DS) lamL[t] = pw[t][CH_L];
}

// ---------------------------------------------------------------------------
// Phase 1: per (16-d group, 16-column group):
//   Y_intra = K_d @ X   (written to d_out)
//   S_inc   = Bm_d @ X  (written to workspace state buffer)
// 16x16x16 fp32 product = 4 chained V_WMMA_F32_16X16X4_F32.
// ---------------------------------------------------------------------------
__global__ __launch_bounds__(256) void s4_phase1(const float* __restrict__ x,
                                                 const float* __restrict__ mats,
                                                 float* __restrict__ y,
                                                 float* __restrict__ Sbuf) {
  __shared__ float Xs[16 * SDX];        // [dl][s][col] : dl*SDX + s*17 + col
  __shared__ float Ms[2 * 16 * SDM];    // K then Bm; later reused as Y/S staging
  int tid = threadIdx.x;
  int dg = blockIdx.x >> 7;             // 64 channel groups
  int cg = blockIdx.x & 127;            // 128 column groups
  int d0 = dg * 16;
  int col0 = cg * 16;
  int b  = col0 / NCH;                  // all 16 cols share one batch index
  int c0 = col0 % NCH;

  // cooperative load of K and Bm for 16 channels (row-padded to 18)
  for (int i = tid; i < 16 * 512; i += 256) {
    int dl = i >> 9;
    int r  = i & 511;
    int which = r >> 8;                 // 0 = K, 1 = Bm
    int e = r & 255;
    int m = e >> 4, k = e & 15;
    Ms[which * 16 * SDM + dl * SDM + m * 18 + k] =
        mats[(size_t)(d0 + dl) * MATS_PER_D + which * 256 + e];
  }
  // cooperative load of X tile: lanes sweep 16 consecutive d (coalesced)
  for (int pass = 0; pass < 16; ++pass) {
    int p  = (tid >> 4) + pass * 16;    // (col,s) pair 0..255
    int cl = p >> 4, s = p & 15;
    int dl = tid & 15;
    int tg = (c0 + cl) * CH_L + s;
    Xs[dl * SDX + s * 17 + cl] = x[((size_t)b * SEQ + tg) * DM + d0 + dl];
  }
  __syncthreads();

  int wave = tid >> 5;
  int lane = tid & 31;
  int half = lane >> 4, lm = lane & 15;
  v8f accY[2], accS[2];
  for (int i = 0; i < 2; ++i) {
    int dl = wave * 2 + i;              // each wave owns 2 channels
    v8f ay = {}; v8f as = {};
    const float* Xd = &Xs[dl * SDX];
    const float* Kd = &Ms[dl * SDM];
    const float* Bd = &Ms[16 * SDM + dl * SDM];
#pragma unroll
    for (int j = 0; j < 4; ++j) {
      int k = 4 * j + 2 * half;         // ISA A/B K-slot mapping for f32 WMMA
      v2f bv; bv.x = Xd[k * 17 + lm];      bv.y = Xd[(k + 1) * 17 + lm];
      v2f aK; aK.x = Kd[lm * 18 + k];      aK.y = Kd[lm * 18 + k + 1];
      v2f aB; aB.x = Bd[lm * 18 + k];      aB.y = Bd[lm * 18 + k + 1];
      ay = __builtin_amdgcn_wmma_f32_16x16x4_f32(false, aK, false, bv, (short)0, ay, false, false);
      as = __builtin_amdgcn_wmma_f32_16x16x4_f32(false, aB, false, bv, (short)0, as, false, false);
    }
    accY[i] = ay; accS[i] = as;
  }
  __syncthreads();
  // stage results over the (now dead) matrix region: [dl][col][row]
  float* Yst = Ms;
  float* Sst = Ms + 16 * SDM;
  for (int i = 0; i < 2; ++i) {
    int dl = wave * 2 + i;
#pragma unroll
    for (int r = 0; r < 8; ++r) {
      int row = r + 8 * half;           // C/D layout: M = r + 8*(lane/16)
      Yst[dl * SDX + lm * 17 + row] = accY[i][r];
      Sst[dl * SDX + lm * 17 + row] = accS[i][r];
    }
  }
  __syncthreads();
  // Y_intra -> d_out, lanes sweep d (coalesced)
  for (int pass = 0; pass < 16; ++pass) {
    int p  = (tid >> 4) + pass * 16;
    int cl = p >> 4, t = p & 15;
    int dl = tid & 15;
    int tg = (c0 + cl) * CH_L + t;
    y[((size_t)b * SEQ + tg) * DM + d0 + dl] = Yst[dl * SDX + cl * 17 + t];
  }
  // S_inc -> workspace [d][col][n], lanes sweep n (coalesced)
  for (int pass = 0; pass < 16; ++pass) {
    int p  = (tid >> 4) + pass * 16;    // (dl, cl) pairs
    int dl = p >> 4, cl = p & 15;
    int n  = tid & 15;
    Sbuf[((size_t)(d0 + dl) * NCOL + b * NCH + c0 + cl) * DS + n] =
        Sst[dl * SDX + cl * 17 + n];
  }
}

// ---------------------------------------------------------------------------
// Phase 2: sequential chunk-state scan, in place:
//   reads inc[c], overwrites with state-entering-chunk-c, h = lamL*h + inc.
// 131072 independent streams, 256 steps each.
// ---------------------------------------------------------------------------
__global__ void s4_phase2(const float* __restrict__ mats, float* __restrict__ Sbuf) {
  int tl = blockIdx.x * blockDim.x + threadIdx.x;   // 0..131071
  int n  = tl & 15;
  int bb = (tl >> 4) & 7;
  int d  = tl >> 7;
  float lam = mats[(size_t)d * MATS_PER_D + 768 + n];
  size_t base = ((size_t)d * NCOL + bb * NCH) * DS + n;
  float h = 0.0f;
  for (int c = 0; c < NCH; ++c) {
    size_t idx = base + (size_t)c * DS;
    float inc = Sbuf[idx];
    Sbuf[idx] = h;
    h = fmaf(lam, h, inc);
  }
}

// ---------------------------------------------------------------------------
// Phase 3: Y += Cm_d @ H  (WMMA again), read-modify-write on d_out.
// ---------------------------------------------------------------------------
__global__ __launch_bounds__(256) void s4_phase3(const float* __restrict__ mats,
                                                 const float* __restrict__ Hbuf,
                                                 float* __restrict__ y) {
  __shared__ float Hs[16 * SDX];        // [dl][n][col]
  __shared__ float Ms[16 * SDM];        // Cm; reused as Y staging
  int tid = threadIdx.x;
  int dg = blockIdx.x >> 7, cg = blockIdx.x & 127;
  int d0 = dg * 16, col0 = cg * 16;
  int b = col0 / NCH, c0 = col0 % NCH;

  for (int i = tid; i < 16 * 256; i += 256) {
    int dl = i >> 8, e = i & 255;
    int m = e >> 4, k = e & 15;
    Ms[dl * SDM + m * 18 + k] = mats[(size_t)(d0 + dl) * MATS_PER_D + 512 + e];
  }
  for (int pass = 0; pass < 16; ++pass) {
    int p  = (tid >> 4) + pass * 16;    // (dl, cl)
    int dl = p >> 4, cl = p & 15;
    int n  = tid & 15;
    Hs[dl * SDX + n * 17 + cl] =
        Hbuf[((size_t)(d0 + dl) * NCOL + b * NCH + c0 + cl) * DS + n];
  }
  __syncthreads();

  int wave = tid >> 5, lane = tid & 31, half = lane >> 4, lm = lane & 15;
  v8f acc[2];
  for (int i = 0; i < 2; ++i) {
    int dl = wave * 2 + i;
    v8f a = {};
    const float* Hd = &Hs[dl * SDX];
    const float* Cd = &Ms[dl * SDM];
#pragma unroll
    for (int j = 0; j < 4; ++j) {
      int k = 4 * j + 2 * half;
      v2f bv; bv.x = Hd[k * 17 + lm];  bv.y = Hd[(k + 1) * 17 + lm];
      v2f ac; ac.x = Cd[lm * 18 + k];  ac.y = Cd[lm * 18 + k + 1];
      a = __builtin_amdgcn_wmma_f32_16x16x4_f32(false, ac, false, bv, (short)0, a, false, false);
    }
    acc[i] = a;
  }
  __syncthreads();
  float* Yst = Ms;
  for (int i = 0; i < 2; ++i) {
    int dl = wave * 2 + i;
#pragma unroll
    for (int r = 0; r < 8; ++r)
      Yst[dl * SDX + lm * 17 + (r + 8 * half)] = acc[i][r];
  }
  __syncthreads();
  for (int pass = 0; pass < 16; ++pass) {
    int p  = (tid >> 4) + pass * 16;
    int cl = p >> 4, t = p & 15;
    int dl = tid & 15;
    size_t oi = ((size_t)b * SEQ + (size_t)(c0 + cl) * CH_L + t) * DM + d0 + dl;
    y[oi] += Yst[dl * SDX + cl * 17 + t];
  }
}

// ---------------------------------------------------------------------------
// Fallback: plain register-resident recurrence, one thread per (b,d).
// Used only if workspace is too small for the chunked-scan path.
// ---------------------------------------------------------------------------
__global__ void s4_fallback(const float* __restrict__ x, const float* __restrict__ A_log,
                            const float* __restrict__ B, const float* __restrict__ C,
                            const float* __restrict__ D, float* __restrict__ y) {
  int tl = blockIdx.x * blockDim.x + threadIdx.x;   // 0..8191
  int d = tl & (DM - 1);
  int b = tl >> 10;
  float lam[DS], Bv[DS], Cv[DS], h[DS];
#pragma unroll
  for (int n = 0; n < DS; ++n) {
    lam[n] = 1.0f - __expf(A_log[d * DS + n]);
    Bv[n] = B[d * DS + n];
    Cv[n] = C[d * DS + n];
    h[n] = 0.0f;
  }
  float Dd = D[d];
  for (int t = 0; t < SEQ; ++t) {
    size_t idx = ((size_t)b * SEQ + t) * DM + d;
    float u = x[idx];
    float acc = u * Dd;
#pragma unroll
    for (int n = 0; n < DS; ++n) {
      h[n] = fmaf(h[n], lam[n], u * Bv[n]);
      acc = fmaf(h[n], Cv[n], acc);
    }
    y[idx] = acc;
  }
}

extern "C" void kernel_launch(void* const* d_in, const int* in_sizes, int n_in,
                              void* d_out, int out_size, void* d_ws, size_t ws_size,
                              hipStream_t stream) {
  const float* x     = (const float*)d_in[0];
  const float* A_log = (const float*)d_in[1];
  const float* B     = (const float*)d_in[2];
  const float* C     = (const float*)d_in[3];
  const float* D     = (const float*)d_in[4];
  float* out = (float*)d_out;

  const size_t mats_floats = (size_t)DM * MATS_PER_D;                 // ~0.8M
  const size_t sbuf_floats = (size_t)DM * NCOL * DS;                  // 32M
  const size_t need_bytes  = (mats_floats + sbuf_floats) * sizeof(float);

  if (ws_size < need_bytes || d_ws == nullptr) {
    s4_fallback<<<(BATCH * DM) / 256, 256, 0, stream>>>(x, A_log, B, C, D, out);
    return;
  }
  float* mats = (float*)d_ws;
  float* Sbuf = mats + mats_floats;

  s4_prep  <<<DM, 32, 0, stream>>>(A_log, B, C, D, mats);
  s4_phase1<<<(DM / 16) * (NCOL / 16), 256, 0, stream>>>(x, mats, out, Sbuf);
  s4_phase2<<<(DM * BATCH * DS) / 256, 256, 0, stream>>>(mats, Sbuf);
  s4_phase3<<<(DM / 16) * (NCOL / 16), 256, 0, stream>>>(mats, Sbuf, out);
}